// DeepSymmetricGCN1dBlock_11751030522223
// MI455X (gfx1250) — compile-verified
//
#include <hip/hip_runtime.h>

// Problem constants (match reference)
#define L_NODES 2048
#define EDGES   16384
#define BB      4
#define NN      8
#define GG      (BB*NN)   // 32
#define CC      256       // Cout of every layer
#define EPS_BN  1e-5f
#define EPB     64        // edges per scatter block

typedef __attribute__((ext_vector_type(2))) float v2f;
typedef __attribute__((ext_vector_type(8))) float v8f;
typedef __attribute__((ext_vector_type(4))) unsigned int u32x4;
typedef __attribute__((ext_vector_type(8))) int i32x8;
typedef __attribute__((ext_vector_type(4))) int i32x4;

#if __has_builtin(__builtin_amdgcn_tensor_load_to_lds)
#define HAVE_TDM 1
#else
#define HAVE_TDM 0
#endif

// ---------------------------------------------------------------- utilities
__global__ void k_fill(float* __restrict__ p, float v, int n) {
  int i = blockIdx.x * blockDim.x + threadIdx.x;
  if (i < n) p[i] = v;
}

// deg[dst] += 1 for every edge (deg pre-initialized to 2.0 for the improved self-loop)
__global__ void k_deg(const int* __restrict__ ei, float* __restrict__ deg) {
  int e = blockIdx.x * blockDim.x + threadIdx.x;
  if (e < EDGES) atomicAdd(&deg[ei[EDGES + e]], 1.0f);
}

// in-place deg -> dis = rsqrt(deg)   (deg >= 2 always)
__global__ void k_dis(float* __restrict__ d) {
  int i = blockIdx.x * blockDim.x + threadIdx.x;
  if (i < L_NODES) d[i] = rsqrtf(d[i]);
}

// selfw[l] = 2*dis[l]^2 ; enorm[e] = dis[src]*dis[dst]
__global__ void k_edge_norm(const int* __restrict__ ei, const float* __restrict__ dis,
                            float* __restrict__ selfw, float* __restrict__ enorm) {
  int i = blockIdx.x * blockDim.x + threadIdx.x;
  if (i < L_NODES) { float dv = dis[i]; selfw[i] = 2.0f * dv * dv; }
  if (i < EDGES)   enorm[i] = dis[ei[i]] * dis[ei[EDGES + i]];
}

// ---------------------------------------------------------------- WMMA GEMM
// HW[g, l, co] = sum_ci X[g, ci, l] * W[ci, co]
// X: [G, Cin, L] (channel-major), W: [Cin, CC], HW: [G, L, CC] (node-major).
//
// One block (8 waves, wave32) owns one (g, 16-wide l-tile) and produces all
// CC=256 couts: each wave computes two 16x16 tiles.  The X tile [Cin x 16]
// (<=16KB) is staged ONCE into LDS by the Tensor Data Mover (2D tile:
// tile_dim0=16 floats contiguous, tile_dim1=Cin rows, row stride = L), so X
// is read from HBM/L2 exactly once instead of NT=16 times.  A-fragments come
// from LDS (lanes 0-15 / 16-31 hit disjoint bank groups -> conflict-free);
// B-fragments (W, <=256KB, L2-hot) from global.
//
// f32 WMMA fragment layout (16x16x4): A lanes 0-15 hold K={0,1} in v0/v1,
// lanes 16-31 hold K={2,3}; B mirrored; C/D: vgpr r holds rows M=r
// (lanes 0-15) and M=r+8 (lanes 16-31), N = lane%16.
__global__ void __launch_bounds__(256)
k_gcn_gemm(const float* __restrict__ X, const float* __restrict__ W,
           float* __restrict__ HW, int Cin) {
  extern __shared__ float xs[];            // [Cin][16] staged X tile (dynamic LDS)
  const int lane = threadIdx.x & 31;
  const int wave = threadIdx.x >> 5;
  const int MT = L_NODES / 16;             // 128 l tiles
  const int g  = blockIdx.x / MT;
  const int lb = (blockIdx.x % MT) * 16;   // l base
  const float* __restrict__ Xg = X + (size_t)g * Cin * L_NODES + lb;

#if HAVE_TDM
  if (wave == 0) {
    // Tensor DMA descriptor (ISA 8.3/8.4): 2D tile load Global -> LDS.
    const unsigned lds_off = (unsigned)__builtin_amdgcn_groupstaticsize(); // dyn-LDS base
    const unsigned long long ga = (unsigned long long)(const void*)Xg;
    u32x4 g0; i32x8 g1; i32x4 g2 = {}; i32x4 g3 = {}; i32x8 g4 = {};
    g0[0] = 1u;                                            // count=1, user descriptor
    g0[1] = lds_off;                                       // lds_addr (bytes)
    g0[2] = (unsigned)(ga & 0xFFFFFFFFu);                  // global_addr[31:0]
    g0[3] = (unsigned)((ga >> 32) & 0x01FFFFFFu)           // global_addr[56:32]
          | (2u << 30);                                    // type = 2 ("image")
    g1[0] = (2 << 16);                                     // data_size = 4 bytes
    g1[1] = (int)((unsigned)(L_NODES & 0xFFFF) << 16);     // tensor_dim0[15:0]
    g1[2] = (int)(((L_NODES >> 16) & 0xFFFF)
          | ((unsigned)(Cin & 0xFFFF) << 16));             // dim0 hi | tensor_dim1 lo
    g1[3] = (int)(((Cin >> 16) & 0xFFFF) | (16u << 16));   // dim1 hi | tile_dim0 = 16
    g1[4] = Cin;                                           // tile_dim1 = Cin (tile_dim2=0)
    g1[5] = L_NODES;                                       // tensor_dim0_stride lo
    g1[6] = 0; g1[7] = 0;                                  // stride hi, dim1_stride
    __builtin_amdgcn_tensor_load_to_lds(g0, g1, g2, g3, g4, 0);
    __builtin_amdgcn_s_wait_tensorcnt((unsigned short)0);
  }
#else
  for (int i = threadIdx.x; i < Cin * 16; i += 256)
    xs[i] = Xg[(size_t)(i >> 4) * L_NODES + (i & 15)];
#endif
  __syncthreads();

  const int half = lane >> 4;              // 0: lanes 0-15, 1: lanes 16-31
  const int mi   = lane & 15;
#pragma unroll
  for (int t = 0; t < 2; ++t) {
    const int cb = (wave + t * 8) * 16;    // cout base: waves cover all 16 tiles
    v8f acc = {};
    for (int k = 0; k < Cin; k += 4) {
      const int k0 = k + (half ? 2 : 0);
      v2f a, b;
      a.x = xs[k0 * 16 + mi];
      a.y = xs[(k0 + 1) * 16 + mi];
      b.x = W[(size_t)k0 * CC + cb + mi];
      b.y = W[(size_t)(k0 + 1) * CC + cb + mi];
      acc = __builtin_amdgcn_wmma_f32_16x16x4_f32(
          /*neg_a=*/false, a, /*neg_b=*/false, b,
          /*c_mod=*/(short)0, acc, /*reuse_a=*/false, /*reuse_b=*/false);
    }
    float* __restrict__ out = HW + ((size_t)g * L_NODES + lb) * CC + cb;
#pragma unroll
    for (int r = 0; r < 8; ++r) {
      const int M = r + half * 8;
      out[(size_t)M * CC + mi] = acc[r];
    }
  }
}

// ---------------------------------------------------------------- aggregation
// AG[g,l,c] = bias[c] + selfw[l] * HW[g,l,c]   (self-loop term, non-atomic init)
__global__ void k_self_init(const float* __restrict__ HW, const float* __restrict__ bias,
                            const float* __restrict__ selfw, float* __restrict__ AG, int G) {
  size_t i   = (size_t)blockIdx.x * blockDim.x + threadIdx.x;
  size_t tot = (size_t)G * L_NODES * CC;
  if (i >= tot) return;
  int c = (int)(i % CC);
  int l = (int)((i / CC) % L_NODES);
  AG[i] = bias[c] + selfw[l] * HW[i];
}

// AG[g,dst,c] += enorm[e] * HW[g,src,c]  — [L,C] layout => 1KB coalesced atomic rows
__global__ void __launch_bounds__(256)
k_scatter(const float* __restrict__ HW, const int* __restrict__ ei,
          const float* __restrict__ enorm, float* __restrict__ AG, int G) {
  int egrp = blockIdx.x % (EDGES / EPB);
  int g    = blockIdx.x / (EDGES / EPB);
  const float* __restrict__ hwg = HW + (size_t)g * L_NODES * CC;
  float* __restrict__ agg       = AG + (size_t)g * L_NODES * CC;
  int c  = threadIdx.x;
  int e0 = egrp * EPB;
  for (int e = e0; e < e0 + EPB; ++e) {
    int s = ei[e], d = ei[EDGES + e];
    float w = enorm[e];
    atomicAdd(&agg[(size_t)d * CC + c], w * hwg[(size_t)s * CC + c]);
  }
}

// ---------------------------------------------------------------- batch-norm
// sums[c] += sum(v), sums[CC+c] += sum(v^2) over all rows (coalesced)
__global__ void __launch_bounds__(256)
k_stats(const float* __restrict__ AG, float* __restrict__ sums, int rows) {
  int c = threadIdx.x;
  float s = 0.f, q = 0.f;
  for (int r = blockIdx.x; r < rows; r += gridDim.x) {
    float v = AG[(size_t)r * CC + c];
    s += v; q += v * v;
  }
  atomicAdd(&sums[c], s);
  atomicAdd(&sums[CC + c], q);
}

__global__ void k_finalize(const float* __restrict__ sums, float inv_n,
                           float* __restrict__ mv) {
  int c = threadIdx.x;
  float m = sums[c] * inv_n;
  float v = sums[CC + c] * inv_n - m * m;
  mv[c]      = m;
  mv[CC + c] = rsqrtf(v + EPS_BN);
}

// ---------------------------------------------------------------- pooling
// XP[b,c,l] = max_n X[b,n,c,l]
__global__ void k_pool(const float* __restrict__ X, float* __restrict__ XP, int Cin) {
  size_t i   = (size_t)blockIdx.x * blockDim.x + threadIdx.x;
  size_t per = (size_t)Cin * L_NODES;
  size_t tot = (size_t)BB * per;
  if (i >= tot) return;
  size_t cl = i % per;
  size_t b  = i / per;
  const float* __restrict__ xb = X + b * NN * per + cl;
  float m = xb[0];
#pragma unroll
  for (int n = 1; n < NN; ++n) m = fmaxf(m, xb[(size_t)n * per]);
  XP[i] = m;
}

// ---------------------------------------------------------------- combine
// OUT[g,c,l] = relu( A1[c]*AG[g,l,c]+B1[c] + A2[c]*AGS[b,l,c]+B2[c] )
// LDS transpose (padded rows -> conflict-free) so the [C,L] store coalesces.
#define LT 32
__global__ void __launch_bounds__(256)
k_combine(const float* __restrict__ AG, const float* __restrict__ AGS,
          const float* __restrict__ mv1, const float* __restrict__ mv2,
          const float* __restrict__ ga, const float* __restrict__ be,
          const float* __restrict__ gs, const float* __restrict__ bs,
          float* __restrict__ OUT) {
  __shared__ float lds[CC * (LT + 1)];
  int g  = blockIdx.x / (L_NODES / LT);
  int lb = (blockIdx.x % (L_NODES / LT)) * LT;
  int b  = g / NN;
  int c  = threadIdx.x;
  float is1 = mv1[CC + c], is2 = mv2[CC + c];
  float A1 = ga[c] * is1, B1 = be[c] - A1 * mv1[c];
  float A2 = gs[c] * is2, B2 = bs[c] - A2 * mv2[c];
  const float* __restrict__ a1 = AG  + ((size_t)g * L_NODES + lb) * CC + c;
  const float* __restrict__ a2 = AGS + ((size_t)b * L_NODES + lb) * CC + c;
#pragma unroll 4
  for (int l = 0; l < LT; ++l) {
    float v = A1 * a1[(size_t)l * CC] + B1 + A2 * a2[(size_t)l * CC] + B2;
    lds[c * (LT + 1) + l] = fmaxf(v, 0.f);
  }
  __syncthreads();
  float* __restrict__ outg = OUT + (size_t)g * CC * L_NODES + lb;
  int lw = threadIdx.x & 31;
  int c0 = threadIdx.x >> 5;
  for (int cc = c0; cc < CC; cc += 8)
    outg[(size_t)cc * L_NODES + lw] = lds[cc * (LT + 1) + lw];
}

// ---------------------------------------------------------------- driver
extern "C" void kernel_launch(void* const* d_in, const int* in_sizes, int n_in,
                              void* d_out, int out_size, void* d_ws, size_t ws_size,
                              hipStream_t stream) {
  const float* x_in = (const float*)d_in[0];
  const int*   ei   = (const int*)d_in[1];
  const float* W [3] = {(const float*)d_in[2],  (const float*)d_in[4],  (const float*)d_in[6]};
  const float* bi[3] = {(const float*)d_in[3],  (const float*)d_in[5],  (const float*)d_in[7]};
  const float* Ws[3] = {(const float*)d_in[8],  (const float*)d_in[10], (const float*)d_in[12]};
  const float* bs[3] = {(const float*)d_in[9],  (const float*)d_in[11], (const float*)d_in[13]};
  const float* ga[3] = {(const float*)d_in[14], (const float*)d_in[16], (const float*)d_in[18]};
  const float* be[3] = {(const float*)d_in[15], (const float*)d_in[17], (const float*)d_in[19]};
  const float* gas[3]= {(const float*)d_in[20], (const float*)d_in[22], (const float*)d_in[24]};
  const float* bes[3]= {(const float*)d_in[21], (const float*)d_in[23], (const float*)d_in[25]};

  char* ws = (char*)d_ws;
  const size_t big = (size_t)GG * L_NODES * CC * sizeof(float);  // 64 MB
  const size_t sym = (size_t)BB * L_NODES * CC * sizeof(float);  //  8 MB
  float* A     = (float*)ws; ws += big;
  float* Bbuf  = (float*)ws; ws += big;
  float* XP    = (float*)ws; ws += sym;
  float* HWS   = (float*)ws; ws += sym;
  float* AGS   = (float*)ws; ws += sym;
  float* dis   = (float*)ws; ws += L_NODES * sizeof(float);
  float* selfw = (float*)ws; ws += L_NODES * sizeof(float);
  float* enorm = (float*)ws; ws += EDGES * sizeof(float);
  float* sums1 = (float*)ws; ws += 2 * CC * sizeof(float);
  float* sums2 = (float*)ws; ws += 2 * CC * sizeof(float);
  float* mv1   = (float*)ws; ws += 2 * CC * sizeof(float);
  float* mv2   = (float*)ws; ws += 2 * CC * sizeof(float);

  // --- graph normalization (once) ---
  k_fill<<<(L_NODES + 255) / 256, 256, 0, stream>>>(dis, 2.0f, L_NODES);
  k_deg <<<(EDGES   + 255) / 256, 256, 0, stream>>>(ei, dis);
  k_dis <<<(L_NODES + 255) / 256, 256, 0, stream>>>(dis);
  k_edge_norm<<<(EDGES + 255) / 256, 256, 0, stream>>>(ei, dis, selfw, enorm);

  const float* xcur = x_in;
  for (int li = 0; li < 3; ++li) {
    const int Cin = (li == 0) ? 128 : 256;
    const size_t shmem = (size_t)Cin * 16 * sizeof(float);  // staged X tile
    float *hw, *agg, *out;
    if (li == 0)      { hw = A;    agg = Bbuf; out = A; }
    else if (li == 1) { hw = Bbuf; agg = A;    out = Bbuf; }
    else              { hw = A;    agg = Bbuf; out = (float*)d_out; }

    // per-element GEMM, then pool (both read xcur BEFORE agg may alias it)
    k_gcn_gemm<<<GG * (L_NODES / 16), 256, shmem, stream>>>(xcur, W[li], hw, Cin);
    k_pool<<<(int)(((size_t)BB * Cin * L_NODES + 255) / 256), 256, 0, stream>>>(xcur, XP, Cin);
    k_gcn_gemm<<<BB * (L_NODES / 16), 256, shmem, stream>>>(XP, Ws[li], HWS, Cin);

    // per-element aggregation + BN stats
    k_self_init<<<(int)((size_t)GG * L_NODES * CC / 256), 256, 0, stream>>>(hw, bi[li], selfw, agg, GG);
    k_scatter<<<GG * (EDGES / EPB), 256, 0, stream>>>(hw, ei, enorm, agg, GG);
    k_fill<<<2, 256, 0, stream>>>(sums1, 0.0f, 2 * CC);
    k_stats<<<256, 256, 0, stream>>>(agg, sums1, GG * L_NODES);
    k_finalize<<<1, 256, 0, stream>>>(sums1, 1.0f / (float)(GG * L_NODES), mv1);

    // symmetric branch aggregation + BN stats
    k_self_init<<<(int)((size_t)BB * L_NODES * CC / 256), 256, 0, stream>>>(HWS, bs[li], selfw, AGS, BB);
    k_scatter<<<BB * (EDGES / EPB), 256, 0, stream>>>(HWS, ei, enorm, AGS, BB);
    k_fill<<<2, 256, 0, stream>>>(sums2, 0.0f, 2 * CC);
    k_stats<<<256, 256, 0, stream>>>(AGS, sums2, BB * L_NODES);
    k_finalize<<<1, 256, 0, stream>>>(sums2, 1.0f / (float)(BB * L_NODES), mv2);

    // fused BN + BN + ReLU + transpose back to [C, L]
    k_combine<<<GG * (L_NODES / LT), 256, 0, stream>>>(agg, AGS, mv1, mv2,
                                                       ga[li], be[li], gas[li], bes[li], out);
    xcur = out;
  }
}